// GNN_44555990729334
// MI455X (gfx1250) — compile-verified
//
#include <hip/hip_runtime.h>
#include <hip/hip_bf16.h>
#include <math.h>

typedef float v2f __attribute__((ext_vector_type(2)));
typedef float v8f __attribute__((ext_vector_type(8)));

// ---------------------------------------------------------------------------
// Degree / normalization kernels
// ---------------------------------------------------------------------------
__global__ void gcn_deg_init(float* __restrict__ deg, int n) {
    int i = blockIdx.x * blockDim.x + threadIdx.x;
    if (i < n) deg[i] = 1.0f;   // self-loop contributes 1 to every node's degree
}

__global__ void gcn_deg_count(const int* __restrict__ col, float* __restrict__ deg, int e) {
    int i = blockIdx.x * blockDim.x + threadIdx.x;
    if (i < e) atomicAdd(&deg[col[i]], 1.0f);
}

__global__ void gcn_deg_rsqrt(float* __restrict__ deg, int n) {
    int i = blockIdx.x * blockDim.x + threadIdx.x;
    if (i < n) deg[i] = rsqrtf(deg[i]);   // deg >= 1 always (self-loops)
}

// ---------------------------------------------------------------------------
// FP32 WMMA GEMM:  C[N x NC] = A[N x K] * B[K x NC], row-major, N % 16 == 0,
// NC == 16 * (blockDim.x / 32).  One wave computes one 16x16 tile, looping K
// in steps of 4 with V_WMMA_F32_16X16X4_F32.
//
// A 16x4 fp32 layout (ISA 7.12.2): lanes 0-15 -> M=lane, VGPR0=K0,VGPR1=K1;
//                                  lanes 16-31 -> M=lane-16, VGPR0=K2,VGPR1=K3.
// B 4x16 mirrors A with N in place of M. C/D: VGPR r -> row r (lanes 0-15)
// or row r+8 (lanes 16-31), col = lane & 15.
// ---------------------------------------------------------------------------
__global__ void gcn_gemm_wmma_f32(const float* __restrict__ A,
                                  const float* __restrict__ B,
                                  float* __restrict__ C,
                                  int K, int NC) {
    const int wave    = threadIdx.x >> 5;
    const int lane    = threadIdx.x & 31;
    const int rowBase = blockIdx.x * 16;
    const int colBase = wave * 16;
    const int mn      = lane & 15;          // M for A, N for B/C
    const int kk      = (lane >> 4) << 1;   // 0 or 2

    const float* aPtr = A + (size_t)(rowBase + mn) * K + kk;
    const float* bPtr = B + (size_t)kk * NC + colBase + mn;

    v8f acc = {};
    for (int k0 = 0; k0 < K; k0 += 4) {
        v2f a, b;
        a.x = aPtr[0];
        a.y = aPtr[1];
        b.x = bPtr[0];
        b.y = bPtr[NC];
        acc = __builtin_amdgcn_wmma_f32_16x16x4_f32(
            /*neg_a=*/false, a, /*neg_b=*/false, b,
            /*c_mod=*/(short)0, acc, /*reuse_a=*/false, /*reuse_b=*/false);
        aPtr += 4;
        bPtr += (size_t)4 * NC;
    }

    float* cPtr = C + (size_t)(rowBase + (lane >> 4) * 8) * NC + colBase + mn;
#pragma unroll
    for (int r = 0; r < 8; ++r) {
        cPtr[(size_t)r * NC] = acc[r];
    }
}

// ---------------------------------------------------------------------------
// agg[n][c] = h[n][c] * dis[n]^2   (self-loop term; fully initializes agg)
// ---------------------------------------------------------------------------
__global__ void gcn_self_seed(const float* __restrict__ h,
                              const float* __restrict__ dis,
                              float* __restrict__ agg,
                              int shift, int total) {
    int i = blockIdx.x * blockDim.x + threadIdx.x;
    if (i >= total) return;
    float d = dis[i >> shift];
    agg[i] = h[i] * d * d;
}

// ---------------------------------------------------------------------------
// Edge scatter-add: one wave per edge.  128-ch: 4 ch/lane (float4);
// 64-ch: 2 ch/lane (float2).  agg[col] += h[row] * dis[row] * dis[col]
// ---------------------------------------------------------------------------
__global__ void gcn_edge_agg128(const float* __restrict__ h,
                                const int* __restrict__ row,
                                const int* __restrict__ col,
                                const float* __restrict__ dis,
                                float* __restrict__ agg, int nEdges) {
    int e    = (int)((blockIdx.x * (size_t)blockDim.x + threadIdx.x) >> 5);
    int lane = threadIdx.x & 31;
    if (e >= nEdges) return;
    int r = row[e], c = col[e];
    float nrm = dis[r] * dis[c];
    const float4 v = ((const float4*)(h + (size_t)r * 128))[lane];
    float* dst = agg + (size_t)c * 128 + lane * 4;
    atomicAdd(dst + 0, v.x * nrm);
    atomicAdd(dst + 1, v.y * nrm);
    atomicAdd(dst + 2, v.z * nrm);
    atomicAdd(dst + 3, v.w * nrm);
}

__global__ void gcn_edge_agg64(const float* __restrict__ h,
                               const int* __restrict__ row,
                               const int* __restrict__ col,
                               const float* __restrict__ dis,
                               float* __restrict__ agg, int nEdges) {
    int e    = (int)((blockIdx.x * (size_t)blockDim.x + threadIdx.x) >> 5);
    int lane = threadIdx.x & 31;
    if (e >= nEdges) return;
    int r = row[e], c = col[e];
    float nrm = dis[r] * dis[c];
    const float2 v = ((const float2*)(h + (size_t)r * 64))[lane];
    float* dst = agg + (size_t)c * 64 + lane * 2;
    atomicAdd(dst + 0, v.x * nrm);
    atomicAdd(dst + 1, v.y * nrm);
}

// ---------------------------------------------------------------------------
// In-place bias + relu
// ---------------------------------------------------------------------------
__global__ void gcn_bias_relu(float* __restrict__ agg,
                              const float* __restrict__ bias,
                              int cmask, int total) {
    int i = blockIdx.x * blockDim.x + threadIdx.x;
    if (i >= total) return;
    agg[i] = fmaxf(agg[i] + bias[i & cmask], 0.0f);
}

// ---------------------------------------------------------------------------
// Per-node log_softmax over 64 channels: one wave per node, 2 ch per lane.
// out[n][c] = z[c] - max - log(sum exp(z - max)),  z = agg2[n] + b2
// ---------------------------------------------------------------------------
__global__ void gcn_logsoftmax64(const float* __restrict__ agg,
                                 const float* __restrict__ bias,
                                 float* __restrict__ out, int nNodes) {
    int node = (int)((blockIdx.x * (size_t)blockDim.x + threadIdx.x) >> 5);
    int lane = threadIdx.x & 31;
    if (node >= nNodes) return;
    const float* z = agg + (size_t)node * 64;
    float z0 = z[lane]      + bias[lane];
    float z1 = z[lane + 32] + bias[lane + 32];

    float m = fmaxf(z0, z1);
#pragma unroll
    for (int off = 16; off > 0; off >>= 1)
        m = fmaxf(m, __shfl_xor(m, off, 32));

    float s = expf(z0 - m) + expf(z1 - m);
#pragma unroll
    for (int off = 16; off > 0; off >>= 1)
        s += __shfl_xor(s, off, 32);

    float lse = logf(s) + m;
    float* o = out + (size_t)node * 64;
    o[lane]      = z0 - lse;
    o[lane + 32] = z1 - lse;
}

// ---------------------------------------------------------------------------
// Launcher
// ---------------------------------------------------------------------------
extern "C" void kernel_launch(void* const* d_in, const int* in_sizes, int n_in,
                              void* d_out, int out_size, void* d_ws, size_t ws_size,
                              hipStream_t stream) {
    const float* x  = (const float*)d_in[0];
    const int*   ei = (const int*)  d_in[1];
    const float* W1 = (const float*)d_in[2];
    const float* b1 = (const float*)d_in[3];
    const float* W2 = (const float*)d_in[4];
    const float* b2 = (const float*)d_in[5];
    float* out = (float*)d_out;

    const int IN = 128, HID = 128, OUT = 64;
    const int N = in_sizes[0] / IN;      // 100000 (multiple of 16)
    const int E = in_sizes[1] / 2;       // 1600000
    const int* rowIdx = ei;              // edge_index[0] = sources
    const int* colIdx = ei + E;          // edge_index[1] = targets

    // Workspace layout (floats):
    //   [0, 131072)                 dis (degree -> rsqrt, N used)
    //   [131072, +N*128)            h   (layer-1 GEMM out; later reused:
    //                                    first N*64 = outp, rest free)
    //   [.., +N*128)                agg1 (layer-1 aggregate -> hrelu in place)
    //   agg2 lives in the back half of the dead h region.
    float* ws   = (float*)d_ws;
    float* dis  = ws;
    float* h    = ws + (1 << 17);
    float* agg1 = h + (size_t)N * HID;
    float* outp = h;                       // N*OUT, reuses dead h region
    float* agg2 = h + (size_t)N * OUT;     // N*OUT, disjoint from outp

    const int T = 256;

    // --- normalization: deg -> dis ---
    gcn_deg_init<<<(N + T - 1) / T, T, 0, stream>>>(dis, N);
    gcn_deg_count<<<(E + T - 1) / T, T, 0, stream>>>(colIdx, dis, E);
    gcn_deg_rsqrt<<<(N + T - 1) / T, T, 0, stream>>>(dis, N);

    // --- layer 1: h = x @ W1 ---
    gcn_gemm_wmma_f32<<<N / 16, 8 * 32, 0, stream>>>(x, W1, h, IN, HID);

    // --- layer 1 aggregation ---
    gcn_self_seed<<<((size_t)N * HID + T - 1) / T, T, 0, stream>>>(h, dis, agg1, 7, N * HID);
    gcn_edge_agg128<<<(E + 7) / 8, T, 0, stream>>>(h, rowIdx, colIdx, dis, agg1, E);
    gcn_bias_relu<<<((size_t)N * HID + T - 1) / T, T, 0, stream>>>(agg1, b1, HID - 1, N * HID);

    // --- layer 2: outp = hrelu @ W2 ---
    gcn_gemm_wmma_f32<<<N / 16, 4 * 32, 0, stream>>>(agg1, W2, outp, HID, OUT);

    // --- layer 2 aggregation ---
    gcn_self_seed<<<((size_t)N * OUT + T - 1) / T, T, 0, stream>>>(outp, dis, agg2, 6, N * OUT);
    gcn_edge_agg64<<<(E + 7) / 8, T, 0, stream>>>(outp, rowIdx, colIdx, dis, agg2, E);

    // --- bias + log_softmax -> d_out ---
    gcn_logsoftmax64<<<(N + 7) / 8, T, 0, stream>>>(agg2, b2, out, N);
}